// vGINEncoder_65111704207431
// MI455X (gfx1250) — compile-verified
//
#include <hip/hip_runtime.h>

// ---------------- problem constants (fixed by the reference) ----------------
#define N_NODES   50000
#define N_EDGES   800000
#define G_GRAPHS  512
#define H_DIM     96
#define DH_DIM    192
#define BN_EPS    1e-5f

typedef float v2f __attribute__((ext_vector_type(2)));
typedef float v8f __attribute__((ext_vector_type(8)));

// ---------------------------------------------------------------------------
__global__ void fill_zero(float* __restrict__ p, int n) {
    int i = blockIdx.x * blockDim.x + threadIdx.x;
    if (i < n) p[i] = 0.f;
}

__global__ void bcast_vn(const float* __restrict__ vn, float* __restrict__ vnf) {
    vnf[(size_t)blockIdx.x * H_DIM + threadIdx.x] = vn[threadIdx.x];
}

// agg[dst[e]] += h[src[e]]  (block = 96 channels, grid = E)
__global__ void scatter_add_edges(const float* __restrict__ h,
                                  const int* __restrict__ src,
                                  const int* __restrict__ dst,
                                  float* __restrict__ agg) {
    int e = blockIdx.x;
    int c = threadIdx.x;
    int s = src[e], d = dst[e];
    atomicAdd(&agg[(size_t)d * H_DIM + c], h[(size_t)s * H_DIM + c]);
}

// out = post + vnf[batch[node]]
__global__ void add_vn(const float* __restrict__ post, const float* __restrict__ vnf,
                       const int* __restrict__ batch, float* __restrict__ out) {
    int node = blockIdx.x, c = threadIdx.x;
    int g = batch[node];
    out[(size_t)node * H_DIM + c] = post[(size_t)node * H_DIM + c] + vnf[(size_t)g * H_DIM + c];
}

// pooled[batch[node]] += h[node]
__global__ void pool_add(const float* __restrict__ h, const int* __restrict__ batch,
                         float* __restrict__ pooled) {
    int node = blockIdx.x, c = threadIdx.x;
    int g = batch[node];
    atomicAdd(&pooled[(size_t)g * H_DIM + c], h[(size_t)node * H_DIM + c]);
}

__global__ void count_nodes(const int* __restrict__ batch, float* __restrict__ counts) {
    int i = blockIdx.x * blockDim.x + threadIdx.x;
    if (i < N_NODES) atomicAdd(&counts[batch[i]], 1.0f);
}

__global__ void mean_div(const float* __restrict__ sums, const float* __restrict__ counts,
                         float* __restrict__ out) {
    int g = blockIdx.x, c = threadIdx.x;
    out[(size_t)g * H_DIM + c] = sums[(size_t)g * H_DIM + c] / fmaxf(counts[g], 1.0f);
}

// colsum/colsumsq -> affine BN params: scale = g*rsqrt(var+eps), shift = b - mu*scale
__global__ void bn_finalize(const float* __restrict__ sum, const float* __restrict__ sumsq,
                            const float* __restrict__ gamma, const float* __restrict__ beta,
                            float inv_n, float* __restrict__ scale, float* __restrict__ shift) {
    int c = threadIdx.x;
    float mu  = sum[c] * inv_n;
    float var = sumsq[c] * inv_n - mu * mu;
    float rs  = rsqrtf(var + BN_EPS);
    float sc  = gamma[c] * rs;
    scale[c] = sc;
    shift[c] = beta[c] - mu * sc;
}

// out = [relu]( h*scale[c] + shift[c] ), block = C channels, grid = rows
template <bool RELU>
__global__ void bn_apply(const float* __restrict__ h, const float* __restrict__ scale,
                         const float* __restrict__ shift, float* __restrict__ out) {
    size_t i = (size_t)blockIdx.x * blockDim.x + threadIdx.x;
    int c = threadIdx.x;
    float v = fmaf(h[i], scale[c], shift[c]);
    if (RELU) v = fmaxf(v, 0.f);
    out[i] = v;
}

// ---------------------------------------------------------------------------
// f32 WMMA GEMM:  C[N,P] = op(A)[N,K] @ W[K,P] + bias
//   op(A): optional (A + A2), optional relu(A*ascale[k] + ashift[k])  (fused BN)
// One wave -> 16 x (16*NT) tile via V_WMMA_F32_16X16X4_F32, 4 waves/block over M.
// All 4 waves share the same weight columns, so the W tile (K x 16*NT, 18-25 KB)
// is staged once per block into LDS (coalesced), then B fragments come from LDS.
// Epilogue: store C, atomically accumulate per-column sum / sum-of-squares for BN.
//
// A-frag (16x4 f32, 2 VGPRs): lane = M (mod 16), reg r holds K = r + 2*(lane>=16)
//   -> regs r=0,1 are contiguous columns: one 8B-aligned v2f load.
// B-frag (4x16 f32, 2 VGPRs): lane = N (mod 16), reg r holds K = r + 2*(lane>=16)
//   -> two ds_load_b32 from LDS rows kk, kk+1 (consecutive lanes = consecutive
//      banks, conflict-free).
// C-frag (16x16 f32, 8 VGPRs): lane = N (mod 16), reg r holds M = r + 8*(lane>=16)
template <int K, int NT, bool ADD2, bool BNA>
__launch_bounds__(128)
__global__ void gemm_wmma_f32(const float* __restrict__ A,  const float* __restrict__ A2,
                              const float* __restrict__ ascale, const float* __restrict__ ashift,
                              const float* __restrict__ W,  const float* __restrict__ bias,
                              float* __restrict__ C,
                              float* __restrict__ colsum, float* __restrict__ colsumsq,
                              int N, int P) {
    static_assert(K % 4 == 0, "K must be a multiple of 4");
    constexpr int TN = 16 * NT;
    __shared__ float Wt[K * TN];

    const int nbase = blockIdx.y * TN;

    // ---- cooperative stage of the W tile (all 128 threads, before any early-out)
    for (int idx = threadIdx.x; idx < K * TN; idx += 128) {
        const int k = idx / TN;
        const int c = idx - k * TN;
        Wt[idx] = W[(size_t)k * P + nbase + c];
    }
    __syncthreads();

    const int wave = threadIdx.x >> 5;
    const int lane = threadIdx.x & 31;
    const int half = lane >> 4;       // 0: lanes 0-15, 1: lanes 16-31
    const int l15  = lane & 15;
    const int tileM = blockIdx.x * 4 + wave;
    if (tileM * 16 >= N) return;      // wave-uniform early-out (EXEC all-ones for WMMA)
    const int m = tileM * 16 + l15;

    v8f acc[NT] = {};

    for (int k = 0; k < K; k += 4) {
        const int kk = k + 2 * half;          // even -> 8B aligned v2f loads
        v2f a = *(const v2f*)(A + (size_t)m * K + kk);
        if (ADD2) {
            v2f a2 = *(const v2f*)(A2 + (size_t)m * K + kk);
            a[0] += a2[0]; a[1] += a2[1];
        }
        if (BNA) {
            v2f sc = *(const v2f*)(ascale + kk);
            v2f sh = *(const v2f*)(ashift + kk);
            a[0] = fmaxf(fmaf(a[0], sc[0], sh[0]), 0.f);
            a[1] = fmaxf(fmaf(a[1], sc[1], sh[1]), 0.f);
        }
#pragma unroll
        for (int t = 0; t < NT; ++t) {
            v2f b;
            b[0] = Wt[(kk    ) * TN + t * 16 + l15];
            b[1] = Wt[(kk + 1) * TN + t * 16 + l15];
            acc[t] = __builtin_amdgcn_wmma_f32_16x16x4_f32(
                false, a, false, b, (short)0, acc[t], false, false);
        }
    }

#pragma unroll
    for (int t = 0; t < NT; ++t) {
        const int col = nbase + t * 16 + l15;
        const float bv = bias[col];
        float s = 0.f, sq = 0.f;
#pragma unroll
        for (int r = 0; r < 8; ++r) {
            const float v = acc[t][r] + bv;
            const int row = tileM * 16 + r + 8 * half;   // all 8 values share column `col`
            C[(size_t)row * P + col] = v;
            s += v; sq += v * v;
        }
        atomicAdd(&colsum[col], s);
        atomicAdd(&colsumsq[col], sq);
    }
}

// ---------------------------------------------------------------------------
extern "C" void kernel_launch(void* const* d_in, const int* in_sizes, int n_in,
                              void* d_out, int out_size, void* d_ws, size_t ws_size,
                              hipStream_t stream) {
    (void)in_sizes; (void)n_in; (void)out_size; (void)ws_size;

    // ---- inputs (setup_inputs order) ----
    const float* x        = (const float*)d_in[0];
    const int*   ei       = (const int*)  d_in[1];
    const int*   batch    = (const int*)  d_in[2];
    const float* vn_embed = (const float*)d_in[4];
    const float* c1_W1 = (const float*)d_in[5];  const float* c1_b1 = (const float*)d_in[6];
    const float* c1_g1 = (const float*)d_in[7];  const float* c1_be1= (const float*)d_in[8];
    const float* c1_W2 = (const float*)d_in[9];  const float* c1_b2 = (const float*)d_in[10];
    const float* bn1_g = (const float*)d_in[11]; const float* bn1_b = (const float*)d_in[12];
    const float* cs_W1 = (const float*)d_in[13]; const float* cs_b1 = (const float*)d_in[14];
    const float* cs_g1 = (const float*)d_in[15]; const float* cs_be1= (const float*)d_in[16];
    const float* cs_W2 = (const float*)d_in[17]; const float* cs_b2 = (const float*)d_in[18];
    const float* bns_g = (const float*)d_in[19]; const float* bns_b = (const float*)d_in[20];
    const float* vm_W1 = (const float*)d_in[21]; const float* vm_b1 = (const float*)d_in[22];
    const float* vm_g1 = (const float*)d_in[23]; const float* vm_be1= (const float*)d_in[24];
    const float* vm_W2 = (const float*)d_in[25]; const float* vm_b2 = (const float*)d_in[26];
    const float* vm_g2 = (const float*)d_in[27]; const float* vm_be2= (const float*)d_in[28];
    const int* src = ei;
    const int* dst = ei + N_EDGES;
    float* out = (float*)d_out;

    // ---- workspace layout (floats) ----
    float* ws = (float*)d_ws;
    size_t o = 0;
    float* agg    = ws + o; o += (size_t)N_NODES * H_DIM;      // 4.8M
    float* tmp    = ws + o; o += (size_t)N_NODES * H_DIM;      // 4.8M
    float* z      = ws + o; o += (size_t)N_NODES * DH_DIM;     // 9.6M
    float* h2     = ws + o; o += (size_t)N_NODES * H_DIM;      // 4.8M
    float* post   = ws + o; o += (size_t)N_NODES * H_DIM;      // 4.8M
    float* vnf    = ws + o; o += (size_t)G_GRAPHS * H_DIM;
    float* pooled = ws + o; o += (size_t)G_GRAPHS * H_DIM;
    float* vmz    = ws + o; o += (size_t)G_GRAPHS * DH_DIM;
    float* vmh    = ws + o; o += (size_t)G_GRAPHS * H_DIM;
    float* counts = ws + o; o += 512;
    float* stats  = ws + o; o += 2048;
    // stats bank A (DH-wide) and bank B (H-wide)
    float* sumA = stats;        float* sqA = stats + 256;
    float* sclA = stats + 512;  float* shfA = stats + 768;
    float* sumB = stats + 1024; float* sqB = stats + 1280;
    float* sclB = stats + 1536; float* shfB = stats + 1792;

    auto zero = [&](float* p, size_t n) {
        fill_zero<<<(unsigned)((n + 255) / 256), 256, 0, stream>>>(p, (int)n);
    };

    const dim3 gemm_blk(128);
    const dim3 g_conv1((N_NODES / 16 + 3) / 4, DH_DIM / 48);   // NT=3
    const dim3 g_conv2((N_NODES / 16 + 3) / 4, H_DIM / 32);    // NT=2
    const dim3 g_vm1(G_GRAPHS / 64, DH_DIM / 48);              // NT=3
    const dim3 g_vm2(G_GRAPHS / 64, H_DIM / 32);               // NT=2
    const float invN = 1.f / (float)N_NODES;
    const float invG = 1.f / (float)G_GRAPHS;

    // virtual-node features: broadcast embedding row
    bcast_vn<<<G_GRAPHS, H_DIM, 0, stream>>>(vn_embed, vnf);

    // ---------------- layer 1 ----------------
    zero(agg, (size_t)N_NODES * H_DIM);
    scatter_add_edges<<<N_EDGES, H_DIM, 0, stream>>>(x, src, dst, agg);
    zero(sumA, 512);
    gemm_wmma_f32<H_DIM, 3, true, false><<<g_conv1, gemm_blk, 0, stream>>>(
        x, agg, nullptr, nullptr, c1_W1, c1_b1, z, sumA, sqA, N_NODES, DH_DIM);
    bn_finalize<<<1, DH_DIM, 0, stream>>>(sumA, sqA, c1_g1, c1_be1, invN, sclA, shfA);
    zero(sumB, 512);
    gemm_wmma_f32<DH_DIM, 2, false, true><<<g_conv2, gemm_blk, 0, stream>>>(
        z, nullptr, sclA, shfA, c1_W2, c1_b2, h2, sumB, sqB, N_NODES, H_DIM);
    bn_finalize<<<1, H_DIM, 0, stream>>>(sumB, sqB, bn1_g, bn1_b, invN, sclB, shfB);
    bn_apply<true><<<N_NODES, H_DIM, 0, stream>>>(h2, sclB, shfB, post);

    // ---------------- layers 2..3 (loop) ----------------
    for (int i = 0; i < 2; ++i) {
        add_vn<<<N_NODES, H_DIM, 0, stream>>>(post, vnf, batch, tmp);
        zero(agg, (size_t)N_NODES * H_DIM);
        scatter_add_edges<<<N_EDGES, H_DIM, 0, stream>>>(tmp, src, dst, agg);
        zero(sumA, 512);
        gemm_wmma_f32<H_DIM, 3, true, false><<<g_conv1, gemm_blk, 0, stream>>>(
            tmp, agg, nullptr, nullptr,
            cs_W1 + (size_t)i * H_DIM * DH_DIM, cs_b1 + (size_t)i * DH_DIM,
            z, sumA, sqA, N_NODES, DH_DIM);
        bn_finalize<<<1, DH_DIM, 0, stream>>>(sumA, sqA,
            cs_g1 + (size_t)i * DH_DIM, cs_be1 + (size_t)i * DH_DIM, invN, sclA, shfA);
        zero(sumB, 512);
        gemm_wmma_f32<DH_DIM, 2, false, true><<<g_conv2, gemm_blk, 0, stream>>>(
            z, nullptr, sclA, shfA,
            cs_W2 + (size_t)i * DH_DIM * H_DIM, cs_b2 + (size_t)i * H_DIM,
            h2, sumB, sqB, N_NODES, H_DIM);
        bn_finalize<<<1, H_DIM, 0, stream>>>(sumB, sqB,
            bns_g + (size_t)i * H_DIM, bns_b + (size_t)i * H_DIM, invN, sclB, shfB);

        if (i == 0) {
            bn_apply<true><<<N_NODES, H_DIM, 0, stream>>>(h2, sclB, shfB, post);
            // virtual-node update: vnf = MLP(globalAddPool(post) + vnf)
            zero(pooled, (size_t)G_GRAPHS * H_DIM);
            pool_add<<<N_NODES, H_DIM, 0, stream>>>(post, batch, pooled);
            zero(sumA, 512);
            gemm_wmma_f32<H_DIM, 3, true, false><<<g_vm1, gemm_blk, 0, stream>>>(
                pooled, vnf, nullptr, nullptr, vm_W1, vm_b1, vmz, sumA, sqA,
                G_GRAPHS, DH_DIM);
            bn_finalize<<<1, DH_DIM, 0, stream>>>(sumA, sqA, vm_g1, vm_be1, invG, sclA, shfA);
            zero(sumB, 512);
            gemm_wmma_f32<DH_DIM, 2, false, true><<<g_vm2, gemm_blk, 0, stream>>>(
                vmz, nullptr, sclA, shfA, vm_W2, vm_b2, vmh, sumB, sqB,
                G_GRAPHS, H_DIM);
            bn_finalize<<<1, H_DIM, 0, stream>>>(sumB, sqB, vm_g2, vm_be2, invG, sclB, shfB);
            bn_apply<true><<<G_GRAPHS, H_DIM, 0, stream>>>(vmh, sclB, shfB, vnf);
        } else {
            bn_apply<false><<<N_NODES, H_DIM, 0, stream>>>(h2, sclB, shfB, post);
        }
    }

    // ---------------- GlobalMeanPool readout ----------------
    zero(pooled, (size_t)G_GRAPHS * H_DIM);
    zero(counts, 512);
    pool_add<<<N_NODES, H_DIM, 0, stream>>>(post, batch, pooled);
    count_nodes<<<(N_NODES + 255) / 256, 256, 0, stream>>>(batch, counts);
    mean_div<<<G_GRAPHS, H_DIM, 0, stream>>>(pooled, counts, out);
}